// ReshapeLayer_146028888173
// MI455X (gfx1250) — compile-verified
//
#include <hip/hip_runtime.h>

// out[n, h*32+i, w*32+j] = x[n, j*32+i, h, w]
// N=256, C=1024 (=32*32), H=W=16. Pure permute: HBM-roofline bound (~22us).
//
// Strategy: per workgroup handle one (n, h-pair) tile.
//  - Async-DMA 1024 channel-chunks (128B each: h in {2hh,2hh+1}, w=0..15)
//    global -> LDS, reordering channels to c' = i*32 + j, chunk stride 36
//    floats (pad) so store-phase j-gathers are only 2-way bank conflicted.
//  - Store phase writes one fully contiguous 128KB output block with
//    coalesced float4 stores.

__global__ __launch_bounds__(256)
void transposed_pixel_shuffle_kernel(const float* __restrict__ x,
                                     float* __restrict__ out) {
    // 1024 chunks * 36 floats = 147456 B = 144 KB LDS
    __shared__ float lds[1024 * 36];

    const int t  = threadIdx.x;
    const int n  = blockIdx.x >> 3;   // 256 images
    const int hh = blockIdx.x & 7;    // 8 h-pairs

    // Base of this tile in input: x + n*C*H*W + hh*2*16  (floats)
    const float* xbase = x + (size_t)n * (1024 * 256) + hh * 32;

    // LDS base byte offset (addrspace(3) offset lives in low 32 bits)
    const unsigned lds_base = (unsigned)(uintptr_t)(&lds[0]);

    // ---- Load phase: global -> LDS via 128-bit async DMA ----
    // 8192 float4s per tile; each thread issues 32 async b128 loads.
    #pragma unroll 4
    for (int it = 0; it < 32; ++it) {
        const int g   = it * 256 + t;   // float4 index in tile [0, 8192)
        const int cp  = g >> 3;         // LDS chunk c' = i*32 + j
        const int sub = g & 7;          // 16B slot within 128B chunk
        const int i   = cp >> 5;
        const int j   = cp & 31;
        const int cin = j * 32 + i;     // source channel c = j*32 + i
        const float* src = xbase + cin * 256 + sub * 4;
        const unsigned ldsoff = lds_base + (unsigned)(cp * 144 + sub * 16);
        asm volatile("global_load_async_to_lds_b128 %0, %1, off"
                     :
                     : "v"(ldsoff), "v"(src)
                     : "memory");
    }
    // Wait for this wave's async deposits, then cross-wave barrier.
    asm volatile("s_wait_asynccnt 0x0" ::: "memory");
    __syncthreads();

    // ---- Store phase: LDS -> global, fully coalesced float4 ----
    // Output tile = contiguous 128KB block: rows r = 64*hh .. 64*hh+63 of the
    // (512 x 512) output image for this n.
    float4* outf4 = (float4*)out + (size_t)n * 65536 + hh * 8192;

    #pragma unroll 4
    for (int it = 0; it < 32; ++it) {
        const int o  = it * 256 + t;    // float4 index in 128KB block
        const int lf = o * 4;           // local float index
        const int rl = lf >> 9;         // local row 0..63 = hl*32 + i
        const int q  = lf & 511;        // column = w*32 + j
        const int hl = rl >> 5;
        const int i  = rl & 31;
        const int w  = q >> 5;
        const int j0 = q & 31;          // float4 spans j0..j0+3 (same w)
        const int base = (i * 32 + j0) * 36 + hl * 16 + w;
        float4 v;
        v.x = lds[base];
        v.y = lds[base + 36];
        v.z = lds[base + 72];
        v.w = lds[base + 108];
        outf4[o] = v;
    }
}

extern "C" void kernel_launch(void* const* d_in, const int* in_sizes, int n_in,
                              void* d_out, int out_size, void* d_ws, size_t ws_size,
                              hipStream_t stream) {
    const float* x = (const float*)d_in[0];
    float* out = (float*)d_out;
    // Grid: N * (H/2) = 256 * 8 = 2048 workgroups, 256 threads each.
    transposed_pixel_shuffle_kernel<<<dim3(2048), dim3(256), 0, stream>>>(x, out);
}